// ExtractOverlappingPatches_29635274342941
// MI455X (gfx1250) — compile-verified
//
#include <hip/hip_runtime.h>
#include <hip/hip_bf16.h>
#include <stdint.h>

// y[b, c*9 + f, h, w] = x[b, c, h+i-1, w+j-1], f = i*3+j, zero padded.
// Pure data movement: ~67MB read (L2-resident), ~604MB streamed write.
// Bandwidth floor @23.3TB/s ~= 29us.

#define B_  16
#define C_  64
#define H_  128
#define W_  128
#define ROWS 8                 // output rows per block
#define TILE_ROWS (ROWS + 2)   // 10 staged input rows
#define TILE_W 136             // 4 pad + 128 + 4 pad (16B-aligned interior)

typedef float v4f __attribute__((ext_vector_type(4)));
typedef int   v4i __attribute__((ext_vector_type(4)));

typedef __attribute__((address_space(1))) v4i* gv4i_ptr;   // global
typedef __attribute__((address_space(3))) v4i* lv4i_ptr;   // LDS

#if defined(__has_builtin)
#  if __has_builtin(__builtin_amdgcn_global_load_async_to_lds_b128)
#    define HAS_ASYNC_LDS 1
#  endif
#endif

__global__ __launch_bounds__(256) void
extract_patches_kernel(const float* __restrict__ x, float* __restrict__ y) {
    __shared__ float tile[TILE_ROWS][TILE_W];

    const int t   = threadIdx.x;
    const int bc  = blockIdx.x;          // b*64 + c
    const int h0  = blockIdx.y * ROWS;   // first output row of this block
    const int b   = bc >> 6;
    const int c   = bc & 63;

    const float* __restrict__ xbase = x + (size_t)bc * (H_ * W_);
    float* xnc = const_cast<float*>(xbase);   // builtin takes non-const ptr

    // ---- Stage 10 input rows (h0-1 .. h0+8) into LDS, interior at col 4 ----
#if HAS_ASYNC_LDS
    for (int idx = t; idx < TILE_ROWS * 32; idx += 256) {
        const int r    = idx >> 5;
        const int lane = idx & 31;
        const int hh   = h0 + r - 1;
        if (hh >= 0 && hh < H_) {
            __builtin_amdgcn_global_load_async_to_lds_b128(
                (gv4i_ptr)(xnc + hh * W_ + lane * 4),
                (lv4i_ptr)&tile[r][4 + lane * 4],
                0, 0);
        } else {
            v4f z = {0.f, 0.f, 0.f, 0.f};
            *(v4f*)&tile[r][4 + lane * 4] = z;
        }
    }
    // zero the 4-column pads on both sides of every staged row
    if (t < TILE_ROWS * 8) {
        const int r = t >> 3;
        const int k = t & 7;
        tile[r][(k < 4) ? k : (TILE_W - 8 + k)] = 0.f;
    }
#  if __has_builtin(__builtin_amdgcn_s_wait_asynccnt)
    __builtin_amdgcn_s_wait_asynccnt(0);
#  else
    asm volatile("s_wait_asynccnt 0" ::: "memory");
#  endif
#else
    for (int idx = t; idx < TILE_ROWS * 32; idx += 256) {
        const int r    = idx >> 5;
        const int lane = idx & 31;
        const int hh   = h0 + r - 1;
        v4f v = {0.f, 0.f, 0.f, 0.f};
        if (hh >= 0 && hh < H_)
            v = *(const v4f*)(xbase + hh * W_ + lane * 4);
        *(v4f*)&tile[r][4 + lane * 4] = v;
    }
    if (t < TILE_ROWS * 8) {
        const int r = t >> 3;
        const int k = t & 7;
        tile[r][(k < 4) ? k : (TILE_W - 8 + k)] = 0.f;
    }
#endif
    __syncthreads();

    // ---- Each thread: one output row (r), 4 consecutive w (w0..w0+3) ----
    const int r    = t >> 5;        // 0..7  (whole wave shares r)
    const int lane = t & 31;
    const int w0   = lane << 2;
    const int h    = h0 + r;

    float* __restrict__ ybase =
        y + (((size_t)b * (C_ * 9) + (size_t)c * 9) * H_ + h) * W_ + w0;

    #pragma unroll
    for (int i = 0; i < 3; ++i) {
        // input row h+i-1 is staged at tile row r+i; tile col = (w + j - 1) + 4
        const float* row = &tile[r + i][0];
        v4f   a  = *(const v4f*)(row + w0 + 4);   // cols w0+4 .. w0+7 (16B aligned)
        float sm = row[w0 + 3];                   // left neighbor
        float sp = row[w0 + 8];                   // right neighbor

        v4f v0 = {sm,  a.x, a.y, a.z};            // j = 0
        v4f v1 = a;                               // j = 1
        v4f v2 = {a.y, a.z, a.w, sp};             // j = 2

        // streaming (non-temporal) stores: don't evict the L2-resident input
        __builtin_nontemporal_store(v0, (v4f*)(ybase + (size_t)(i * 3 + 0) * (H_ * W_)));
        __builtin_nontemporal_store(v1, (v4f*)(ybase + (size_t)(i * 3 + 1) * (H_ * W_)));
        __builtin_nontemporal_store(v2, (v4f*)(ybase + (size_t)(i * 3 + 2) * (H_ * W_)));
    }
}

extern "C" void kernel_launch(void* const* d_in, const int* in_sizes, int n_in,
                              void* d_out, int out_size, void* d_ws, size_t ws_size,
                              hipStream_t stream) {
    (void)in_sizes; (void)n_in; (void)out_size; (void)d_ws; (void)ws_size;
    const float* x = (const float*)d_in[0];
    float*       y = (float*)d_out;

    dim3 grid(B_ * C_, H_ / ROWS, 1);   // 1024 x 16 blocks
    dim3 block(256, 1, 1);              // 8 waves (wave32)
    extract_patches_kernel<<<grid, block, 0, stream>>>(x, y);
}